// GATChainLayer_82686710383220
// MI455X (gfx1250) — compile-verified
//
#include <hip/hip_runtime.h>
#include <hip/hip_bf16.h>

// ---------------------------------------------------------------------------
// GATChainLayer for MI455X (gfx1250, wave32).
// Roofline: GEMMs ~11 GFLOP (negligible on WMMA); traffic ~1 GB (edge
// gathers/atomics + 128MB out) -> bandwidth bound, ~45us floor @ 23.3 TB/s.
// Strategy: f32 WMMA (16x16x4, confirmed lowering) for both GEMMs with
// LDS-staged tiles (coalesced b128 global loads, conflict-free ds reads),
// residual GEMM seeds d_out, edge phase = wave-per-edge kernels with f32
// atomics into d_out.
// ---------------------------------------------------------------------------

#define NND 32768      // nodes
#define FD  128        // input features (K of GEMMs)
#define CD  4          // chains
#define HD  8          // heads
#define DD  32         // head dim
#define DMD 256        // d_model = HD*DD
#define ED  65536      // bond edges
#define BD  8192       // bridge edges
#define ETD (ED + 2*BD)   // 81920 total edges
#define OUTD 1024      // CD*HD*DD

typedef __attribute__((ext_vector_type(2)))  float v2f;
typedef __attribute__((ext_vector_type(8)))  float v8f;

// monotone float<->uint map for atomicMax over signed floats
__device__ __forceinline__ unsigned f2ord(float f) {
  unsigned b = __float_as_uint(f);
  return (b & 0x80000000u) ? ~b : (b | 0x80000000u);
}
__device__ __forceinline__ float ord2f(unsigned u) {
  unsigned b = (u & 0x80000000u) ? (u ^ 0x80000000u) : ~u;
  return __uint_as_float(b);
}

// ---------------------------------------------------------------------------
// sinusoidal positional encoding pe[CD][DMD]
__global__ void pe_kernel(float* __restrict__ pe) {
  int i = blockIdx.x * blockDim.x + threadIdx.x;
  if (i >= CD * DMD) return;
  int c = i / DMD, j = i % DMD;
  int je = j & ~1;
  float div = __powf(10000.0f, (float)je / (float)DMD);
  float ang = (float)c / div;
  pe[i] = (j & 1) ? __cosf(ang) : __sinf(ang);
}

// ---------------------------------------------------------------------------
// Out[N,M] = X[N,FD] @ W[M,FD]^T  via V_WMMA_F32_16X16X4_F32 (wave32).
// Block = 256 threads = 8 waves; block tile = 128 rows x 32 cols.
// Each wave owns a 16x32 output strip (2 accumulators, shared A frag).
// X/W tiles staged in LDS (k-tile 64): coalesced float4 global loads,
// LDS row stride 68 floats -> frag reads hit banks 4*l16+kk (conflict-free).
__global__ __launch_bounds__(256) void gemm_xwT(const float* __restrict__ X,
                                                const float* __restrict__ W,
                                                float* __restrict__ Out,
                                                int M) {
  constexpr int KT = 64, LDK = KT + 4;           // padded LDS stride
  __shared__ float Xs[128 * LDK];                // 34,816 B
  __shared__ float Ws[32 * LDK];                 //  8,704 B
  const int t    = threadIdx.x;
  const int lane = t & 31;
  const int wave = t >> 5;
  const int half = lane >> 4;                    // 0: lanes 0-15, 1: 16-31
  const int l16  = lane & 15;
  const int rowBase0 = blockIdx.x * 128;
  const int colBase0 = blockIdx.y * 32;

  v8f acc0 = {}, acc1 = {};
  for (int kt = 0; kt < FD; kt += KT) {
    __syncthreads();                             // protect LDS reuse
    // stage X tile: 128 rows x 64 k = 2048 float4, 8 per thread (coalesced)
#pragma unroll
    for (int i = 0; i < 8; ++i) {
      const int id = t + i * 256;                // 0..2047
      const int r = id >> 4, kq = (id & 15) * 4; // 16 float4 per row
      *(float4*)&Xs[r * LDK + kq] =
          *(const float4*)&X[(size_t)(rowBase0 + r) * FD + kt + kq];
    }
    // stage W tile: 32 rows x 64 k = 512 float4, 2 per thread (coalesced)
#pragma unroll
    for (int i = 0; i < 2; ++i) {
      const int id = t + i * 256;                // 0..511
      const int r = id >> 4, kq = (id & 15) * 4;
      *(float4*)&Ws[r * LDK + kq] =
          *(const float4*)&W[(size_t)(colBase0 + r) * FD + kt + kq];
    }
    __syncthreads();

    const float* xrow  = &Xs[(wave * 16 + l16) * LDK];
    const float* wrow0 = &Ws[l16 * LDK];
    const float* wrow1 = &Ws[(16 + l16) * LDK];
#pragma unroll
    for (int k0 = 0; k0 < KT; k0 += 4) {
      // f32 A 16x4 layout: lanes0-15 K={k0,k0+1}, lanes16-31 K={k0+2,k0+3}
      const int kk = k0 + 2 * half;
      v2f a, b0, b1;
      a.x  = xrow[kk];  a.y  = xrow[kk + 1];
      b0.x = wrow0[kk]; b0.y = wrow0[kk + 1];
      b1.x = wrow1[kk]; b1.y = wrow1[kk + 1];
      acc0 = __builtin_amdgcn_wmma_f32_16x16x4_f32(false, a, false, b0,
                                                   (short)0, acc0, false, false);
      acc1 = __builtin_amdgcn_wmma_f32_16x16x4_f32(false, a, false, b1,
                                                   (short)0, acc1, false, false);
    }
  }
  // D layout: VGPR v -> row (v + 8*half), col l16
  const int rowBase = rowBase0 + wave * 16;
#pragma unroll
  for (int v = 0; v < 8; ++v) {
    const size_t ro = (size_t)(rowBase + v + 8 * half) * M + colBase0;
    Out[ro + l16]      = acc0[v];
    Out[ro + 16 + l16] = acc1[v];
  }
}

// ---------------------------------------------------------------------------
// scores[e, j] = sum_d double_attn[j,d] * edge_attn[e,j,d]; track global max.
// One wave per edge, lane = d.
__global__ __launch_bounds__(256) void score_kernel(
    const float* __restrict__ xl, const float* __restrict__ pe,
    const float* __restrict__ attnw, const int* __restrict__ eidx,
    const int* __restrict__ bidx, float* __restrict__ scores,
    unsigned* __restrict__ gmax_u) {
  const int gwave = (int)((blockIdx.x * blockDim.x + threadIdx.x) >> 5);
  const int lane = threadIdx.x & 31;
  if (gwave >= ETD) return;
  const int e = gwave;

  int mode, na, nb;
  if (e < ED)            { mode = 0; na = eidx[e];            nb = eidx[ED + e]; }
  else if (e < ED + BD)  { int i = e - ED;      mode = 1; na = bidx[i]; nb = bidx[BD + i]; }
  else                   { int i = e - ED - BD; mode = 2; na = bidx[i]; nb = bidx[BD + i]; }

  float wavemax = -3.4e38f;
#pragma unroll 4
  for (int j = 0; j < 32; ++j) {
    const int c = j >> 3, h = j & 7;
    const int idx = h * DD + lane;
    int p, q, cq;
    bool zero;
    if (mode == 0)      { p = na; q = nb; cq = c;                 zero = false;    }
    else if (mode == 1) { p = nb; q = na; cq = (c < 3) ? c + 1 : 3; zero = (c == 3); }  // src2dst
    else                { p = na; q = nb; cq = (c > 0) ? c - 1 : 0; zero = (c == 0); }  // dst2src
    float v = xl[(size_t)p * DMD + idx] + pe[c * DMD + idx] +
              xl[(size_t)q * DMD + idx] + pe[cq * DMD + idx];
    v = (v >= 0.0f) ? v : 0.2f * v;           // leaky 0.2
    if (zero) v = 0.0f;
    float s = v * attnw[j * DD + lane];
    // butterfly reduce over d (wave32)
    for (int off = 16; off; off >>= 1) s += __shfl_xor(s, off, 32);
    if (lane == 0) scores[(size_t)e * 32 + j] = s;
    wavemax = fmaxf(wavemax, s);
  }
  if (lane == 0) atomicMax(gmax_u, f2ord(wavemax));
}

// ---------------------------------------------------------------------------
// exp (global-max stabilized, in place) + denom segment-sum by full_dst
__global__ __launch_bounds__(256) void expden_kernel(
    float* __restrict__ scores, const unsigned* __restrict__ gmax_u,
    const int* __restrict__ eidx, const int* __restrict__ bidx,
    float* __restrict__ denom) {
  const int t = blockIdx.x * blockDim.x + threadIdx.x;
  if (t >= ETD * 32) return;
  const int e = t >> 5, j = t & 31;
  int dst;
  if (e < ED)           dst = eidx[ED + e];
  else if (e < ED + BD) dst = bidx[BD + (e - ED)];
  else                  dst = bidx[e - ED - BD];
  const float gmax = ord2f(*gmax_u);
  const float v = __expf(scores[t] - gmax);
  scores[t] = v;                               // reuse as exp_s
  atomicAdd(&denom[(size_t)dst * 32 + j], v);
}

// ---------------------------------------------------------------------------
// proj scatter: out[dst] += xr[src_bond[full_src[e]]] * alpha (faithful bug:
// full_src < N < E so x_src[full_src] == xr[src_bond[full_src]]).
// One wave per edge, lane = d.
__global__ __launch_bounds__(256) void scatter_kernel(
    const float* __restrict__ xl, const float* __restrict__ pe,
    const float* __restrict__ exps, const float* __restrict__ denom,
    const int* __restrict__ eidx, const int* __restrict__ bidx,
    float* __restrict__ out) {
  const int gwave = (int)((blockIdx.x * blockDim.x + threadIdx.x) >> 5);
  const int lane = threadIdx.x & 31;
  if (gwave >= ETD) return;
  const int e = gwave;

  int fsrc, fdst;
  if (e < ED)           { fsrc = eidx[e];            fdst = eidx[ED + e];      }
  else if (e < ED + BD) { int i = e - ED;      fsrc = bidx[i];      fdst = bidx[BD + i]; }
  else                  { int i = e - ED - BD; fsrc = bidx[BD + i]; fdst = bidx[i];      }
  const int node2 = eidx[fsrc];                // the double-indirection bug

#pragma unroll 4
  for (int j = 0; j < 32; ++j) {
    const int c = j >> 3, h = j & 7;
    const int idx = h * DD + lane;
    const float alpha = exps[(size_t)e * 32 + j] /
                        (denom[(size_t)fdst * 32 + j] + 1e-16f);
    const float val = (xl[(size_t)node2 * DMD + idx] + pe[c * DMD + idx]) * alpha;
    atomicAdd(&out[(size_t)fdst * OUTD + j * DD + lane], val);
  }
}

// ---------------------------------------------------------------------------
// out = prelu(out + bias), in place (out was seeded with residual GEMM)
__global__ __launch_bounds__(256) void final_kernel(
    float* __restrict__ out, const float* __restrict__ bias,
    const float* __restrict__ prelu_w) {
  const size_t i = (size_t)blockIdx.x * blockDim.x + threadIdx.x;
  if (i >= (size_t)NND * OUTD) return;
  const float w = *prelu_w;
  float v = out[i] + bias[i & (OUTD - 1)];
  out[i] = (v >= 0.0f) ? v : w * v;
}

// ---------------------------------------------------------------------------
extern "C" void kernel_launch(void* const* d_in, const int* in_sizes, int n_in,
                              void* d_out, int out_size, void* d_ws, size_t ws_size,
                              hipStream_t stream) {
  const float* x      = (const float*)d_in[0];
  const int*   eidx   = (const int*)  d_in[1];   // [2,E]
  const int*   bidx   = (const int*)  d_in[2];   // [2,B]
  const float* W_in   = (const float*)d_in[3];   // [256,128]
  const float* dattn  = (const float*)d_in[4];   // [1,32,32]
  const float* W_res  = (const float*)d_in[5];   // [1024,128]
  const float* bias   = (const float*)d_in[6];   // [1024]
  const float* prelu  = (const float*)d_in[7];   // [1]
  float*       out    = (float*)d_out;           // [N,1024]

  // workspace layout (~48.2 MB)
  float*    xl     = (float*)d_ws;                         // N*DMD
  float*    scores = xl + (size_t)NND * DMD;               // ETD*32 (reused as exp_s)
  float*    denom  = scores + (size_t)ETD * 32;            // N*32
  float*    pe     = denom + (size_t)NND * 32;             // CD*DMD
  unsigned* gmax   = (unsigned*)(pe + CD * DMD);           // 1

  hipMemsetAsync(denom, 0, (size_t)NND * 32 * sizeof(float), stream);
  hipMemsetAsync(gmax, 0, sizeof(unsigned), stream);

  pe_kernel<<<dim3((CD * DMD + 255) / 256), dim3(256), 0, stream>>>(pe);

  // xl = x @ W_in^T  (no PE; PE added at gather time)
  gemm_xwT<<<dim3(NND / 128, DMD / 32), dim3(256), 0, stream>>>(x, W_in, xl, DMD);
  // seed out with residual: out = x @ W_res^T
  gemm_xwT<<<dim3(NND / 128, OUTD / 32), dim3(256), 0, stream>>>(x, W_res, out, OUTD);

  const int edgeBlocks = ETD / 8;                          // 8 waves/block
  score_kernel<<<dim3(edgeBlocks), dim3(256), 0, stream>>>(xl, pe, dattn, eidx,
                                                           bidx, scores, gmax);
  expden_kernel<<<dim3((ETD * 32) / 256), dim3(256), 0, stream>>>(scores, gmax,
                                                                  eidx, bidx, denom);
  scatter_kernel<<<dim3(edgeBlocks), dim3(256), 0, stream>>>(xl, pe, scores,
                                                             denom, eidx, bidx, out);
  final_kernel<<<dim3((NND * (size_t)OUTD) / 256), dim3(256), 0, stream>>>(out, bias,
                                                                           prelu);
}